// DataDependentAlibi_11201274708513
// MI455X (gfx1250) — compile-verified
//
#include <hip/hip_runtime.h>
#include <hip/hip_bf16.h>
#include <math.h>

typedef __attribute__((ext_vector_type(2))) float v2f;
typedef __attribute__((ext_vector_type(4))) float f4;
typedef __attribute__((ext_vector_type(8))) float v8f;

__device__ __forceinline__ float log_sigmoid(float z) {
    // jax.nn.log_sigmoid(z) = min(z,0) - log1p(exp(-|z|))  (numerically stable)
    return fminf(z, 0.0f) - log1pf(__expf(-fabsf(z)));
}

// ---------------------------------------------------------------------------
// Stage 1: logits = x @ W^T (+bias), logsigmoid, via V_WMMA_F32_16X16X4_F32.
// One wave (32 lanes) computes a 16(row) x 16(head) tile, K-loop over D=1024.
// A-frag (16x4 f32): lanes 0-15 -> M=lane, {VGPR0,VGPR1} = K {0,1};
//                    lanes 16-31 -> same M, K {2,3}.
// B-frag (4x16 f32): lanes 0-15 -> N=lane with K {0,1}; lanes 16-31 K {2,3}.
// C/D (16x16 f32): VGPR r -> M = r + 8*(lane>=16), N = lane&15.
// fg_raw stored as [(b*H + h)*N + n].
// ---------------------------------------------------------------------------
__global__ __launch_bounds__(32) void k_proj_logsig(
    const float* __restrict__ x, const float* __restrict__ W,
    const float* __restrict__ bias, float* __restrict__ fg,
    int N, int D, int H)
{
    const int  lane  = threadIdx.x;          // 0..31 (wave32)
    const long row0  = (long)blockIdx.x * 16; // flattened b*N + n tile base
    const int  m     = lane & 15;
    const int  koff  = (lane >> 4) << 1;     // 0 for lanes 0-15, 2 for 16-31

    const float* __restrict__ xa = x + (row0 + m) * (long)D + koff; // A rows
    const float* __restrict__ wb = W + (long)m * D + koff;          // B cols (head m)

    v8f acc = {};
    for (int k = 0; k < D; k += 4) {
        v2f a, bb;
        a.x  = xa[k];  a.y  = xa[k + 1];
        bb.x = wb[k];  bb.y = wb[k + 1];
        // 8 args: (neg_a, A, neg_b, B, c_mod, C, reuse_a, reuse_b)
        acc = __builtin_amdgcn_wmma_f32_16x16x4_f32(
            false, a, false, bb, (short)0, acc, false, false);
    }

    const int   h     = lane & 15;
    const int   mo    = (lane >> 4) << 3;    // +8 rows for the upper half-wave
    const float bh    = bias[h];
    const int   batch = (int)(row0 / N);
    const int   nbase = (int)(row0 % N);
    float* __restrict__ orow = fg + ((long)batch * H + h) * (long)N;

#pragma unroll
    for (int r = 0; r < 8; ++r) {
        float z = acc[r] + bh;
        orow[nbase + mo + r] = log_sigmoid(z);
    }
}

// ---------------------------------------------------------------------------
// Stage 2: in-place inclusive cumsum along n for each of the B*H rows.
// One 256-thread block per row; 8 elems/thread + LDS Hillis-Steele scan.
// Assumes N == 2048 (256 * 8).
// ---------------------------------------------------------------------------
__global__ __launch_bounds__(256) void k_cumsum(float* __restrict__ fg, int N)
{
    __shared__ float ssum[256];
    const int t = threadIdx.x;
    float* __restrict__ p = fg + (size_t)blockIdx.x * N + (size_t)t * 8;

    float v[8];
    float run = 0.0f;
#pragma unroll
    for (int i = 0; i < 8; ++i) { v[i] = p[i]; run += v[i]; v[i] = run; }

    ssum[t] = run;
    __syncthreads();
    for (int off = 1; off < 256; off <<= 1) {
        float tv = (t >= off) ? ssum[t - off] : 0.0f;
        __syncthreads();
        ssum[t] += tv;
        __syncthreads();
    }
    const float base = (t > 0) ? ssum[t - 1] : 0.0f;

#pragma unroll
    for (int i = 0; i < 8; ++i) p[i] = v[i] + base;
}

// ---------------------------------------------------------------------------
// Stage 3: out[bh][i][j] = fg[bh][i] - fg[bh][j].  Store-bandwidth bound
// (512 MB). One block per (bh,i) row; each thread writes 2 x b128 with
// non-temporal hint so the 512 MB stream doesn't evict the hot fg table.
// ---------------------------------------------------------------------------
__global__ __launch_bounds__(256) void k_pairwise(
    const float* __restrict__ fg, float* __restrict__ out, int N)
{
    const long row = blockIdx.x;            // bh*N + i
    const long bh  = row / N;
    const int  i   = (int)(row - bh * N);

    const float* __restrict__ fr = fg + bh * (long)N;
    const float fi = fr[i];
    f4* __restrict__ orow = (f4*)(out + row * (long)N);
    const f4* __restrict__ fj = (const f4*)fr;

    const int t = threadIdx.x;
#pragma unroll
    for (int c = 0; c < 2; ++c) {
        const int idx4 = t + c * 256;       // float4 index, 512 per row
        f4 g = fj[idx4];
        f4 r;
        r.x = fi - g.x; r.y = fi - g.y; r.z = fi - g.z; r.w = fi - g.w;
        __builtin_nontemporal_store(r, orow + idx4);
    }
}

extern "C" void kernel_launch(void* const* d_in, const int* in_sizes, int n_in,
                              void* d_out, int out_size, void* d_ws, size_t ws_size,
                              hipStream_t stream)
{
    const float* x    = (const float*)d_in[0];   // [B, N, D]
    const float* W    = (const float*)d_in[1];   // [H, D]
    const float* bias = (const float*)d_in[2];   // [H]
    float* out = (float*)d_out;                  // [B, H, N, N]
    float* fg  = (float*)d_ws;                   // [B*H, N] scratch (256 KB)

    const int  H    = in_sizes[2];                         // 16
    const int  D    = in_sizes[1] / H;                     // 1024
    const long rows = (long)in_sizes[0] / D;               // B*N = 4096
    const int  N    = (int)((long)out_size / (rows * H));  // 2048
    const int  B    = (int)(rows / N);                     // 2

    // Stage 1: WMMA projection + logsigmoid -> fg_raw
    k_proj_logsig<<<(int)(rows / 16), 32, 0, stream>>>(x, W, bias, fg, N, D, H);
    // Stage 2: per-(b,h) cumsum in place
    k_cumsum<<<B * H, 256, 0, stream>>>(fg, N);
    // Stage 3: pairwise broadcast-subtract, NT b128 stores
    k_pairwise<<<(unsigned)(B * (long)H * N), 256, 0, stream>>>(fg, out, N);
}